// node_prompt_layer_feature_weighted_sum_21534966022300
// MI455X (gfx1250) — compile-verified
//
#include <hip/hip_runtime.h>
#include <math.h>

typedef __attribute__((ext_vector_type(2))) float v2f;
typedef __attribute__((ext_vector_type(8))) float v8f;

#define FEAT 64

// Non-returning native f32 atomic add (global_atomic_add_f32, no CAS loop).
__device__ __forceinline__ void atomic_fadd(float* p, float v) {
  unsafeAtomicAdd(p, v);
}

// ---------------------------------------------------------------------------
// Phase 1: emb = elu(x * w) via V_WMMA_F32_16X16X4_F32 against diag(w).
// One wave32 computes a 16x64 node tile. For output column block j (16 cols),
// only k-blocks 4j..4j+3 of the diagonal contribute -> 4 WMMAs per block,
// 16 WMMAs per tile, exact f32.
//
// Layouts (CDNA5 ISA 7.12.2):
//   A (16x4 f32, 2 VGPRs): lanes 0-15: (K0,K1), lanes 16-31: (K2,K3), M=lane%16
//   B (4x16 f32, 2 VGPRs): VGPR0 rows {0,2}, VGPR1 rows {1,3}; N=lane%16,
//                          lane half selects row +2
//   C/D (16x16 f32, 8 VGPRs): VGPR r -> row r (lanes 0-15) / row r+8 (16-31)
// ---------------------------------------------------------------------------
__global__ void elu_emb_wmma_kernel(const float* __restrict__ x,
                                    const float* __restrict__ w,
                                    float* __restrict__ emb,
                                    int ntiles) {
  const int wave = (int)((blockIdx.x * blockDim.x + threadIdx.x) >> 5);
  if (wave >= ntiles) return;              // wave-uniform: EXEC stays all-1s
  const int lane = (int)(threadIdx.x & 31);
  const int half = lane >> 4;              // 0 or 1
  const int l16  = lane & 15;
  const int n0   = wave << 4;              // first node row of tile
  const float* xrow = x + (size_t)(n0 + l16) * FEAT;

#pragma unroll
  for (int j = 0; j < 4; ++j) {            // output column blocks (16 cols each)
    const int cbase = j << 4;
    const float wc = w[cbase + l16];       // w for this lane's column
    v8f acc = {};
#pragma unroll
    for (int q = 0; q < 4; ++q) {          // contributing k-blocks of diag(w)
      const int kb = (j << 2) + q;
      const int k0 = (kb << 2) + (half << 1);      // lane's first absolute K
      const float2 av = *(const float2*)(xrow + k0);
      v2f a; a.x = av.x; a.y = av.y;
      // B[r][c] = w[16j+c] iff c == 4q + r, else 0. This lane holds rows
      // r0 = 2*half (vgpr0 slot) and r0+1 (vgpr1 slot) at column c = l16.
      const int r0 = half << 1;
      v2f b;
      b.x = (l16 == ((q << 2) + r0))     ? wc : 0.0f;
      b.y = (l16 == ((q << 2) + r0 + 1)) ? wc : 0.0f;
      acc = __builtin_amdgcn_wmma_f32_16x16x4_f32(
          /*neg_a=*/false, a, /*neg_b=*/false, b,
          /*c_mod=*/(short)0, acc, /*reuse_a=*/false, /*reuse_b=*/false);
    }
    // ELU epilogue + store D tile: VGPR r -> row n0 + 8*half + r, col cbase+l16
#pragma unroll
    for (int r = 0; r < 8; ++r) {
      float v = acc[r];
      v = (v > 0.0f) ? v : expm1f(v);
      emb[(size_t)(n0 + (half << 3) + r) * FEAT + cbase + l16] = v;
    }
  }
}

// Scalar tail for node counts not divisible by 16 (not hit for N=100000).
__global__ void elu_emb_tail_kernel(const float* __restrict__ x,
                                    const float* __restrict__ w,
                                    float* __restrict__ emb,
                                    int start_node, int n_nodes) {
  const int total = (n_nodes - start_node) * FEAT;
  const int i = (int)(blockIdx.x * blockDim.x + threadIdx.x);
  if (i >= total) return;
  const int node = start_node + i / FEAT;
  const int c = i % FEAT;
  float v = x[(size_t)node * FEAT + c] * w[c];
  emb[(size_t)node * FEAT + c] = (v > 0.0f) ? v : expm1f(v);
}

// ---------------------------------------------------------------------------
// Zero the output (harness poisons d_out).
// ---------------------------------------------------------------------------
__global__ void zero_kernel(float* __restrict__ p, int n) {
  const int i = (int)(blockIdx.x * blockDim.x + threadIdx.x);
  if (i < n) p[i] = 0.0f;
}

// ---------------------------------------------------------------------------
// Phase 2: scatter-add. 16 lanes per edge; each lane gathers float4 from the
// (L2-resident) emb table and issues 4 non-returning global_atomic_add_f32.
// ---------------------------------------------------------------------------
__global__ void scatter_add_kernel(const float* __restrict__ emb,
                                   const int* __restrict__ src,
                                   const int* __restrict__ dst,
                                   float* __restrict__ out,
                                   int n_edges) {
  const long long t = (long long)blockIdx.x * blockDim.x + threadIdx.x;
  const long long e = t >> 4;
  if (e >= n_edges) return;
  const int c = (int)(t & 15) << 2;                 // 4 floats per lane
  const int s = src[e];
  const int d = dst[e];
  const float4 v = *(const float4*)(emb + (size_t)s * FEAT + c);
  float* o = out + (size_t)d * FEAT + c;
  atomic_fadd(o + 0, v.x);
  atomic_fadd(o + 1, v.y);
  atomic_fadd(o + 2, v.z);
  atomic_fadd(o + 3, v.w);
}

// Fallback if workspace can't hold emb: recompute elu(x*w) per edge inline.
__global__ void fused_scatter_kernel(const float* __restrict__ x,
                                     const float* __restrict__ w,
                                     const int* __restrict__ src,
                                     const int* __restrict__ dst,
                                     float* __restrict__ out,
                                     int n_edges) {
  const long long t = (long long)blockIdx.x * blockDim.x + threadIdx.x;
  const long long e = t >> 4;
  if (e >= n_edges) return;
  const int c = (int)(t & 15) << 2;
  const int s = src[e];
  const int d = dst[e];
  const float4 xv = *(const float4*)(x + (size_t)s * FEAT + c);
  const float4 wv = *(const float4*)(w + c);
  float4 v;
  v.x = xv.x * wv.x; v.x = (v.x > 0.0f) ? v.x : expm1f(v.x);
  v.y = xv.y * wv.y; v.y = (v.y > 0.0f) ? v.y : expm1f(v.y);
  v.z = xv.z * wv.z; v.z = (v.z > 0.0f) ? v.z : expm1f(v.z);
  v.w = xv.w * wv.w; v.w = (v.w > 0.0f) ? v.w : expm1f(v.w);
  float* o = out + (size_t)d * FEAT + c;
  atomic_fadd(o + 0, v.x);
  atomic_fadd(o + 1, v.y);
  atomic_fadd(o + 2, v.z);
  atomic_fadd(o + 3, v.w);
}

extern "C" void kernel_launch(void* const* d_in, const int* in_sizes, int n_in,
                              void* d_out, int out_size, void* d_ws, size_t ws_size,
                              hipStream_t stream) {
  const float* x  = (const float*)d_in[0];   // [N, 64]
  const float* w  = (const float*)d_in[1];   // [1, 64]
  const int* src  = (const int*)d_in[2];     // [E]
  const int* dst  = (const int*)d_in[3];     // [E]
  float* out      = (float*)d_out;           // [N, 64]

  const int n_nodes = in_sizes[0] / FEAT;
  const int n_edges = in_sizes[2];

  // 1) zero output (d_out is poisoned by the harness)
  zero_kernel<<<(out_size + 255) / 256, 256, 0, stream>>>(out, out_size);

  const size_t emb_bytes = (size_t)n_nodes * FEAT * sizeof(float);
  const long long scatter_threads = (long long)n_edges * 16;
  const int scatter_blocks = (int)((scatter_threads + 255) / 256);

  if (ws_size >= emb_bytes) {
    float* emb = (float*)d_ws;
    // 2) emb = elu(x*w) via WMMA against diag(w)
    const int ntiles = n_nodes >> 4;
    if (ntiles > 0) {
      const int threads = ntiles * 32;
      elu_emb_wmma_kernel<<<(threads + 255) / 256, 256, 0, stream>>>(x, w, emb, ntiles);
    }
    const int start_tail = ntiles << 4;
    if (start_tail < n_nodes) {
      const int tail = (n_nodes - start_tail) * FEAT;
      elu_emb_tail_kernel<<<(tail + 255) / 256, 256, 0, stream>>>(x, w, emb, start_tail, n_nodes);
    }
    // 3) edge scatter: gather emb[src] (L2-resident), atomic-add into out[dst]
    scatter_add_kernel<<<scatter_blocks, 256, 0, stream>>>(emb, src, dst, out, n_edges);
  } else {
    fused_scatter_kernel<<<scatter_blocks, 256, 0, stream>>>(x, w, src, dst, out, n_edges);
  }
}